// Walker_17927193494330
// MI455X (gfx1250) — compile-verified
//
#include <hip/hip_runtime.h>
#include <stdint.h>

// Random-walk feature kernel for MI455X (gfx1250, wave32).
// All node/edge indices provably fit in 32 bits (array sizes arrive as int):
// the gather chain runs on 32-bit index math and 4-byte loads of the low
// dwords of the int64 inputs (little-endian). Only `choices` values (<2^62)
// stay 64-bit, and adj_bits is probed one dword at a time.
// Explicit warm-up (steps 0..6, validity-guarded) / steady-state (guard-free)
// split keeps the hot loop tight.

template<bool WARM>
__device__ __forceinline__ void walk_step(
    int st, int i, int n, unsigned words, int steps,
    const int*       __restrict__ adj_nodes32,
    const int*       __restrict__ adj_offset32,
    const int*       __restrict__ degrees32,
    const int*       __restrict__ node_id32,
    const unsigned*  __restrict__ adj_bits32,
    const long long* __restrict__ choices,
    float* __restrict__ out_nodes,
    float* __restrict__ out_edges,
    unsigned (&win)[8], unsigned (&idm)[8], unsigned (&edm)[7])
{
    const unsigned cur = win[7];
    const unsigned old = win[6];
    const unsigned long long ch = (unsigned long long)choices[(size_t)st * n + i];

    // Software-pipeline: start pulling next step's (coalesced) choices row
    // before this step's HBM gathers (global_prefetch_b8).
    if (st + 1 < steps)
        __builtin_prefetch(&choices[(size_t)(st + 1) * n + i], 0, 1);

    const unsigned deg = (unsigned)degrees32[2 * cur];
    const unsigned off = (unsigned)adj_offset32[2 * cur];

    // Fast path: power-of-two degree -> mask instead of 64-bit urem.
    unsigned eidx = ((deg & (deg - 1u)) == 0u)
                        ? (unsigned)(ch & (unsigned long long)(deg - 1u))
                        : (unsigned)(ch % (unsigned long long)deg);
    unsigned chosen = off + eidx;
    unsigned nw = (unsigned)adj_nodes32[2 * chosen];

    // Non-backtracking retry (rare, per-lane divergent branch).
    if ((nw == old) && (!WARM || st > 0)) {
        const unsigned nbdeg = (deg == 1u) ? deg : (deg - 1u);
        const unsigned chm = ((nbdeg & (nbdeg - 1u)) == 0u)
                                 ? (unsigned)(ch & (unsigned long long)(nbdeg - 1u))
                                 : (unsigned)(ch % (unsigned long long)nbdeg);
        unsigned a = eidx + 1u + chm;   // < 2*deg
        if (a >= deg) a -= deg;
        chosen = off + a;
        nw = (unsigned)adj_nodes32[2 * chosen];
    }

    // Overlap next step's metadata fetch with this step's bitmap gathers.
    __builtin_prefetch(&degrees32[2 * nw], 0, 1);
    __builtin_prefetch(&adj_offset32[2 * nw], 0, 1);

    const unsigned cid  = (unsigned)node_id32[2 * nw];
    const unsigned word = cid / 63u;            // 32-bit magic-multiply
    const unsigned bitp = cid - word * 63u;     // 0..62
    const unsigned dsel = bitp >> 5;            // which dword holds the bit
    const unsigned bsh  = bitp & 31u;

    const int vstart = WARM ? (8 - (st + 1)) : 0;   // folds to 0 in steady state
    const unsigned tb = 1u << (st + 1);

#pragma unroll
    for (int p = 0; p < 8; ++p) {
        if (p >= vstart && win[p] == nw) idm[p] |= tb;
    }
#pragma unroll
    for (int p = 0; p < 7; ++p) {
        if (p >= vstart) {
            // 4-byte probe of the relevant dword of adj_bits[win[p]][word]
            const unsigned w32 = adj_bits32[2u * (win[p] * words + word) + dsel];
            if ((w32 >> bsh) & 1u) edm[p] |= tb;
        }
    }

    out_nodes[st + 1] = (float)nw;      // v_cvt_f32_u32, exact for index values
    out_edges[st]     = (float)chosen;

#pragma unroll
    for (int p = 0; p < 7; ++p) win[p] = win[p + 1];
    win[7] = nw;
}

__global__ __launch_bounds__(256)
void walker_step_kernel(const int*       __restrict__ adj_nodes32,
                        const int*       __restrict__ adj_offset32,
                        const int*       __restrict__ degrees32,
                        const int*       __restrict__ node_id32,
                        const unsigned*  __restrict__ adj_bits32,
                        const long long* __restrict__ choices,
                        const int*       __restrict__ start32,
                        const float*     __restrict__ feat_scale,
                        float*           __restrict__ out,    // full d_out
                        unsigned*        __restrict__ masks,  // SoA: [15][n]
                        int n, unsigned words, int steps, int direct_x)
{
    const int i = blockIdx.x * blockDim.x + threadIdx.x;
    if (i >= n) return;

    const float scale = feat_scale[0];

    unsigned win[8];
#pragma unroll
    for (int p = 0; p < 8; ++p) win[p] = 0u;
    const unsigned s0 = (unsigned)start32[2 * i];
    win[7] = s0;

    float* out_nodes = out + (size_t)i * (steps + 1);
    float* out_edges = out + (size_t)n * (steps + 1) + (size_t)i * steps;
    float* out_x     = out + (size_t)n * (steps + 1) + (size_t)n * steps;

    out_nodes[0] = (float)s0;

    unsigned idm[8];
    unsigned edm[7];
#pragma unroll
    for (int p = 0; p < 8; ++p) idm[p] = 0u;
#pragma unroll
    for (int p = 0; p < 7; ++p) edm[p] = 0u;

    // Warm-up: validity window still filling (st+1 < 8).
    const int warm = (steps < 7) ? steps : 7;
    int st = 0;
    for (; st < warm; ++st)
        walk_step<true>(st, i, n, words, steps, adj_nodes32, adj_offset32,
                        degrees32, node_id32, adj_bits32, choices,
                        out_nodes, out_edges, win, idm, edm);
    // Steady state: guard-free, 7 unconditional bitmap gathers per step.
    for (; st < steps; ++st)
        walk_step<false>(st, i, n, words, steps, adj_nodes32, adj_offset32,
                         degrees32, node_id32, adj_bits32, choices,
                         out_nodes, out_edges, win, idm, edm);

    if (direct_x) {
        // Fallback: write walk_x directly (uncoalesced but correct).
        float* xrow = out_x + (size_t)i * 15 * (steps + 1);
#pragma unroll
        for (int p = 0; p < 8; ++p)
            for (int t = 0; t <= steps; ++t)
                xrow[p * (steps + 1) + t] = ((idm[p] >> t) & 1u) ? scale : 0.0f;
#pragma unroll
        for (int p = 0; p < 7; ++p)
            for (int t = 0; t <= steps; ++t)
                xrow[(8 + p) * (steps + 1) + t] = ((edm[p] >> t) & 1u) ? scale : 0.0f;
    } else {
        // SoA layout -> all 15 mask stores are lane-coalesced.
#pragma unroll
        for (int p = 0; p < 8; ++p) masks[(size_t)p * n + i] = idm[p];
#pragma unroll
        for (int p = 0; p < 7; ++p) masks[(size_t)(8 + p) * n + i] = edm[p];
    }
}

// Phase 2: one block per walker expands 15 bitmasks into 255 contiguous
// floats -> fully coalesced stores for the dominant 51MB output stream.
__global__ __launch_bounds__(256)
void walker_x_kernel(const unsigned* __restrict__ masks,   // SoA: [15][n]
                     const float*    __restrict__ feat_scale,
                     float*          __restrict__ out_x,
                     int n, int steps1)   // steps + 1
{
    const int w = blockIdx.x;
    const int j = threadIdx.x;
    const int total = 15 * steps1;      // 255 for steps=16
    if (j >= total) return;
    const int p = j / steps1;
    const int t = j - p * steps1;
    const unsigned m = masks[(size_t)p * n + w];   // per-block broadcast read
    const float scale = feat_scale[0];
    out_x[(size_t)w * total + j] = ((m >> t) & 1u) ? scale : 0.0f;
}

extern "C" void kernel_launch(void* const* d_in, const int* in_sizes, int n_in,
                              void* d_out, int out_size, void* d_ws, size_t ws_size,
                              hipStream_t stream)
{
    (void)n_in; (void)out_size;

    const int*       adj_nodes32  = (const int*)d_in[0];
    const int*       adj_offset32 = (const int*)d_in[1];
    const int*       degrees32    = (const int*)d_in[2];
    const int*       node_id32    = (const int*)d_in[3];
    const unsigned*  adj_bits32   = (const unsigned*)d_in[4];
    const long long* choices      = (const long long*)d_in[5];
    const int*       start32      = (const int*)d_in[6];
    const float*     feat_scale   = (const float*)d_in[7];
    float* out = (float*)d_out;

    const int n = in_sizes[6];                           // number of walkers
    const unsigned words = (unsigned)(in_sizes[4] / n);  // adj_bits row width
    const int steps = in_sizes[5] / n;                   // walk length

    const size_t mask_bytes = (size_t)n * 15 * sizeof(unsigned);
    const bool two_phase = (ws_size >= mask_bytes);
    unsigned* masks = (unsigned*)d_ws;

    const int block = 256;                               // 8 wave32 per block
    const int grid = (n + block - 1) / block;
    walker_step_kernel<<<grid, block, 0, stream>>>(
        adj_nodes32, adj_offset32, degrees32, node_id32, adj_bits32, choices,
        start32, feat_scale, out, masks, n, words, steps, two_phase ? 0 : 1);

    if (two_phase) {
        float* out_x = out + (size_t)n * (steps + 1) + (size_t)n * steps;
        walker_x_kernel<<<n, 256, 0, stream>>>(masks, feat_scale, out_x, n, steps + 1);
    }
}